// RescaleProbMask_22316650070834
// MI455X (gfx1250) — compile-verified
//
#include <hip/hip_runtime.h>

// RescaleProbMask for x:(32,256,256,16) f32.
// Pass 1: p[h,w] = mean over (batch, channel) via V_WMMA_F32_16X16X4_F32 row-sums.
// Pass 2: elementwise rescale, nontemporal stores to keep x resident in 192MB L2.

typedef float v2f __attribute__((ext_vector_type(2)));
typedef float v4f __attribute__((ext_vector_type(4)));
typedef float v8f __attribute__((ext_vector_type(8)));

#define ALPHA_C   0.25f
#define HW_N      65536           // 256*256 spatial positions
#define PLANE_N   (256 * 256 * 16) // elements per batch image
#define BATCH_N   32

// One wave per 16 consecutive spatial positions (flat hw index). Lane L:
//   m = L & 15  -> position within tile (A-matrix row)
//   half = L>>4 -> which channel-octet this lane supplies
// Per WMMA, row m's 4 k-slots come from lane m (2 vals) + lane m+16 (2 vals).
// With B = ones(4x16): D[m][*] = sum of those 4 values. Four accumulator
// chains c0..c3 cover channel quartets {0,1,8,9},{2,3,10,11},{4,5,12,13},
// {6,7,14,15} — a disjoint partition of all 16 channels — summed over all
// 32 batches, then combined. Full f32 precision throughout.
__global__ __launch_bounds__(256) void rpm_mean_wmma(const float* __restrict__ x,
                                                     float* __restrict__ p) {
  const int wave = blockIdx.x * (blockDim.x >> 5) + (threadIdx.x >> 5);
  const int lane = threadIdx.x & 31;
  const int m    = lane & 15;
  const int half = lane >> 4;

  // tile's first element: position (wave*16) * 16 channels
  const float* src = x + (size_t)wave * 256 + m * 16 + half * 8;

  v8f c0 = {}; v8f c1 = {}; v8f c2 = {}; v8f c3 = {};
  const v2f ones = {1.0f, 1.0f};

#pragma unroll 4
  for (int n = 0; n < BATCH_N; ++n) {
    const float* pn = src + (size_t)n * PLANE_N;
    v4f a = *(const v4f*)pn;        // channels half*8 + 0..3
    v4f b = *(const v4f*)(pn + 4);  // channels half*8 + 4..7
    v2f a0 = {a.x, a.y};
    v2f a1 = {a.z, a.w};
    v2f a2 = {b.x, b.y};
    v2f a3 = {b.z, b.w};
    c0 = __builtin_amdgcn_wmma_f32_16x16x4_f32(false, a0, false, ones, (short)0, c0, false, false);
    c1 = __builtin_amdgcn_wmma_f32_16x16x4_f32(false, a1, false, ones, (short)0, c1, false, false);
    c2 = __builtin_amdgcn_wmma_f32_16x16x4_f32(false, a2, false, ones, (short)0, c2, false, false);
    c3 = __builtin_amdgcn_wmma_f32_16x16x4_f32(false, a3, false, ones, (short)0, c3, false, false);
  }
  v8f c = (c0 + c1) + (c2 + c3);

  // D layout (32-bit C/D 16x16): lane 0 VGPR r = D[r][0]; lane 16 VGPR r = D[8+r][0].
  if (m == 0) {
    const int mbase = half * 8;
#pragma unroll
    for (int r = 0; r < 8; ++r)
      p[wave * 16 + mbase + r] = c[r] * (1.0f / 512.0f);
  }
}

// One float4 per thread; flat element index e = q*4; position = e/16;
// hw = position & 65535 (since each batch image has exactly 65536 positions).
__global__ __launch_bounds__(256) void rpm_apply(const float* __restrict__ x,
                                                 const float* __restrict__ p,
                                                 float* __restrict__ out) {
  const size_t q  = (size_t)blockIdx.x * blockDim.x + threadIdx.x;
  const int    hw = (int)((q >> 2) & (HW_N - 1));

  const float pv   = p[hw];
  const float mul  = ALPHA_C / pv;                  // exact f32 divide
  const float beta = (1.0f - ALPHA_C) / (1.0f - pv);
  const bool  le   = pv >= ALPHA_C;

  v4f xv = *(const v4f*)(x + q * 4);
  v4f r;
#pragma unroll
  for (int i = 0; i < 4; ++i) {
    const float hi = xv[i] * mul;
    const float lo = 1.0f - beta * (1.0f - xv[i]);
    r[i] = le ? hi : lo;
  }
  // Non-temporal: don't let the 128MB output stream evict x from L2 —
  // pass 2's x reads should hit the 192MB L2 populated by pass 1.
  __builtin_nontemporal_store(r, (v4f*)(out + q * 4));
}

extern "C" void kernel_launch(void* const* d_in, const int* in_sizes, int n_in,
                              void* d_out, int out_size, void* d_ws, size_t ws_size,
                              hipStream_t stream) {
  const float* x   = (const float*)d_in[0];
  float*       out = (float*)d_out;
  float*       p   = (float*)d_ws;  // 65536 floats = 256 KB scratch, rewritten every call

  // Pass 1: 65536 positions / 16 per wave = 4096 waves = 512 blocks x 8 waves.
  rpm_mean_wmma<<<512, 256, 0, stream>>>(x, p);

  // Pass 2: 33,554,432 elements / 4 per thread = 8,388,608 threads.
  rpm_apply<<<32768, 256, 0, stream>>>(x, p, out);
}